// KeypointLoss_33234456937091
// MI455X (gfx1250) — compile-verified
//
#include <hip/hip_runtime.h>

#define KEYS   7
#define NCLASS 11
#define S_DIM  5
#define B_DIM  16
#define H_DIM  128
#define W_DIM  128
#define HW     (H_DIM * W_DIM)        // 16384
#define CH_TOT (KEYS + NCLASS)        // 18
#define SBC    (S_DIM * B_DIM * NCLASS) // 880
#define SB     (S_DIM * B_DIM)        // 80

typedef float f4  __attribute__((ext_vector_type(4)));
typedef float v2f __attribute__((ext_vector_type(2)));
typedef float v8f __attribute__((ext_vector_type(8)));

// ---------------------------------------------------------------------------
// Wave32 sum-reduction through V_WMMA_F32_16X16X4_F32 (ones-matrix trick).
// WMMA #1: A vgpr0 = x (A(m,0)=x_m from lanes 0-15, A(m,2)=x_{m+16}), B = ones
//          -> D(m,n) = x_m + x_{m+16}; per-lane sum of the 8 D vgprs yields the
//          half-wave subtotal (lanes 0-15: rows 0-7; lanes 16-31: rows 8-15).
// WMMA #2: feed the subtotal back as A -> every D element = full 32-lane sum.
// ---------------------------------------------------------------------------
__device__ __forceinline__ float wave_reduce_sum(float x) {
#if defined(__gfx1250__) && __has_builtin(__builtin_amdgcn_wmma_f32_16x16x4_f32)
  v2f a;  a[0] = x;   a[1] = 0.0f;
  v2f b;  b[0] = 1.0f; b[1] = 1.0f;     // B(k,n) = 1 regardless of layout
  v8f c = {};
  c = __builtin_amdgcn_wmma_f32_16x16x4_f32(false, a, false, b, (short)0, c,
                                            false, false);
  float s = ((c[0] + c[1]) + (c[2] + c[3])) + ((c[4] + c[5]) + (c[6] + c[7]));
  v2f a2; a2[0] = s;  a2[1] = 0.0f;
  v8f c2 = {};
  c2 = __builtin_amdgcn_wmma_f32_16x16x4_f32(false, a2, false, b, (short)0, c2,
                                             false, false);
  return c2[0];
#else
  for (int off = 16; off > 0; off >>= 1) x += __shfl_xor(x, off, 32);
  return x;
#endif
}

// ---------------------------------------------------------------------------
// Stage 1: one block per (s,b,c). Streams the pred_hm channel and heatmap
// channel once (b128 non-temporal loads), accumulating sum((p-h)^2) and the
// minimum flat index whose heatmap value is 1.0 (== jnp.argmax for 0/1 data,
// with "no one present" encoded as INT_MAX -> mask 0).
// ---------------------------------------------------------------------------
__global__ __launch_bounds__(256) void keypoint_stage1(
    const float* __restrict__ preds,   // (S,B,18,H,W)
    const float* __restrict__ hm,      // (S,B,11,H,W)
    float* __restrict__ heatpart,      // (SBC)
    int*   __restrict__ idxbuf)        // (SBC)
{
  const int blk = blockIdx.x;          // = sb*NCLASS + c
  const int c   = blk % NCLASS;
  const int sb  = blk / NCLASS;
  const int tid = threadIdx.x;

  const f4* __restrict__ predv =
      (const f4*)(preds + ((size_t)sb * CH_TOT + (KEYS + c)) * (size_t)HW);
  const f4* __restrict__ hmv =
      (const f4*)(hm + (size_t)blk * (size_t)HW);

  float acc  = 0.0f;
  int   pmin = 0x7FFFFFFF;

#pragma unroll
  for (int it = 0; it < HW / 4 / 256; ++it) {   // 16 iterations
    const int i = tid + it * 256;               // float4 index
    f4 p = __builtin_nontemporal_load(predv + i);
    f4 h = __builtin_nontemporal_load(hmv + i);
#pragma unroll
    for (int j = 0; j < 4; ++j) {
      float d = p[j] - h[j];
      acc = fmaf(d, d, acc);
      if (h[j] == 1.0f) pmin = min(pmin, 4 * i + j);
    }
  }

  // wave-level sum via WMMA, block-level combine + index min via LDS
  __shared__ float wsum[8];
  __shared__ int   sidx[256];
  const float wtot = wave_reduce_sum(acc);
  const int wave = tid >> 5, lane = tid & 31;
  if (lane == 0) wsum[wave] = wtot;
  sidx[tid] = pmin;
  __syncthreads();
#pragma unroll
  for (int off = 128; off > 0; off >>= 1) {
    if (tid < off) sidx[tid] = min(sidx[tid], sidx[tid + off]);
    __syncthreads();
  }
  if (tid == 0) {
    float t = ((wsum[0] + wsum[1]) + (wsum[2] + wsum[3])) +
              ((wsum[4] + wsum[5]) + (wsum[6] + wsum[7]));
    heatpart[blk] = t;
    idxbuf[blk]   = sidx[0];
  }
}

// ---------------------------------------------------------------------------
// Stage 2: one block per (s,b). Threads 0..76 handle (c,k) gather-loss terms,
// threads 0..10 also fold the per-channel heat partials. Tiny kernel: 6160
// scattered loads total.
// ---------------------------------------------------------------------------
__global__ __launch_bounds__(128) void keypoint_stage2(
    const float* __restrict__ preds,    // (S,B,18,H,W)
    const float* __restrict__ labels,   // (B,11,7)
    const float* __restrict__ heatpart, // (SBC)
    const int*   __restrict__ idxbuf,   // (SBC)
    float* __restrict__ out)            // [0..79]=heat_loss.T, [80..159]=label_loss.T
{
  const int sb = blockIdx.x;            // s*B + b
  const int s  = sb / B_DIM;
  const int b  = sb % B_DIM;
  const int t  = threadIdx.x;

  __shared__ float scls[128];
  __shared__ float sheat[128];

  float cv = 0.0f, hv = 0.0f;
  if (t < NCLASS * KEYS) {
    const int c = t / KEYS, k = t % KEYS;
    const int iv = idxbuf[sb * NCLASS + c];
    if (iv < HW) {                      // mask: a 1.0 peak exists
      float g = preds[((size_t)sb * CH_TOT + k) * (size_t)HW + iv];
      float l = labels[(b * NCLASS + c) * KEYS + k];
      float d = g - l;
      cv = d * d;
    }
  }
  if (t < NCLASS) hv = heatpart[sb * NCLASS + t];

  scls[t]  = cv;
  sheat[t] = hv;
  __syncthreads();
#pragma unroll
  for (int off = 64; off > 0; off >>= 1) {
    if (t < off) { scls[t] += scls[t + off]; sheat[t] += sheat[t + off]; }
    __syncthreads();
  }
  if (t == 0) {
    out[b * S_DIM + s]      = sheat[0];   // heat_loss.T
    out[SB + b * S_DIM + s] = scls[0];    // label_loss.T
  }
}

extern "C" void kernel_launch(void* const* d_in, const int* in_sizes, int n_in,
                              void* d_out, int out_size, void* d_ws, size_t ws_size,
                              hipStream_t stream) {
  const float* preds  = (const float*)d_in[0];   // combined_preds (S,B,18,H,W)
  const float* hm     = (const float*)d_in[1];   // heatmaps       (S,B,11,H,W)
  const float* labels = (const float*)d_in[2];   // labels         (B,11,7)

  float* heatpart = (float*)d_ws;                          // 880 floats
  int*   idxbuf   = (int*)((char*)d_ws + SBC * sizeof(float)); // 880 ints

  keypoint_stage1<<<SBC, 256, 0, stream>>>(preds, hm, heatpart, idxbuf);
  keypoint_stage2<<<SB, 128, 0, stream>>>(preds, labels, heatpart, idxbuf,
                                          (float*)d_out);
}